// EdgePromptBaseline_4501125726343
// MI455X (gfx1250) — compile-verified
//
#include <hip/hip_runtime.h>

// ---------------------------------------------------------------------------
// CDNA5 (gfx1250) EdgePrompt-GIN forward.
//  - edge scatter: wave-per-edge float4 gather + global f32 atomics (L2)
//  - GEMMs (N x 128 @ 128 x 128): v_wmma_f32_16x16x4_f32 with LDS tiles
//    pre-swizzled into fragment-native layout (1x ds_load_b64 per WMMA).
// ---------------------------------------------------------------------------

typedef __attribute__((ext_vector_type(2))) float v2f;
typedef __attribute__((ext_vector_type(8))) float v8f;

static constexpr int HD = 128;  // hidden dim

// ----------------------------- utility ------------------------------------
__global__ void zero_f32_kernel(float* __restrict__ p, long long count) {
  long long i = (long long)blockIdx.x * blockDim.x + threadIdx.x;
  long long stride = (long long)gridDim.x * blockDim.x;
  for (; i < count; i += stride) p[i] = 0.0f;
}

// ------------------------- edge message passing ----------------------------
// Layer 0: D = 12, thread per edge.
__global__ void edge_scatter12_kernel(const float* __restrict__ x,
                                      const int* __restrict__ src,
                                      const int* __restrict__ dst,
                                      const float* __restrict__ p0,
                                      float* __restrict__ agg, int nEdges) {
  int e = blockIdx.x * blockDim.x + threadIdx.x;
  if (e >= nEdges) return;
  int s = src[e], d = dst[e];
  const float* xs = x + (long long)s * 12;
  float* out = agg + (long long)d * 12;
#pragma unroll
  for (int k = 0; k < 12; ++k) atomicAdd(out + k, xs[k] + p0[k]);
}

// Layers 1..4: D = 128, one wave (32 lanes) per edge, float4 per lane.
__global__ void edge_scatter128_kernel(const float* __restrict__ h,
                                       const int* __restrict__ src,
                                       const int* __restrict__ dst,
                                       const float* __restrict__ p,
                                       float* __restrict__ agg, int nEdges) {
  int lane = threadIdx.x & 31;
  int e = blockIdx.x * (blockDim.x >> 5) + (threadIdx.x >> 5);
  if (e >= nEdges) return;
  int s = src[e], d = dst[e];
  float4 v = ((const float4*)(h + (long long)s * HD))[lane];
  float4 pp = ((const float4*)p)[lane];
  float* out = agg + (long long)d * HD + lane * 4;
  atomicAdd(out + 0, v.x + pp.x);
  atomicAdd(out + 1, v.y + pp.y);
  atomicAdd(out + 2, v.z + pp.z);
  atomicAdd(out + 3, v.w + pp.w);
}

// ------------------- layer-0 first MLP (K = 12, VALU) ----------------------
__global__ __launch_bounds__(128) void mlp0_kernel(
    const float* __restrict__ x, const float* __restrict__ agg0,
    const float* __restrict__ W1, const float* __restrict__ b1,
    float* __restrict__ t, int n) {
  __shared__ float Wsh[12 * HD];
  __shared__ float zr[16][12];
  const int tid = threadIdx.x;  // 128 threads: one output column each
  for (int k = tid; k < 12 * HD; k += 128) Wsh[k] = W1[k];
  const int i0 = blockIdx.x * 16;
  for (int idx = tid; idx < 16 * 12; idx += 128) {
    int r = idx / 12, k = idx - r * 12;
    int i = i0 + r;
    zr[r][k] = (i < n) ? (x[(long long)i * 12 + k] + agg0[(long long)i * 12 + k]) : 0.0f;
  }
  __syncthreads();
  const float bj = b1[tid];
  for (int r = 0; r < 16; ++r) {
    int i = i0 + r;
    if (i >= n) break;
    float acc = bj;
#pragma unroll
    for (int k = 0; k < 12; ++k) acc += zr[r][k] * Wsh[k * HD + tid];
    t[(long long)i * HD + tid] = fmaxf(acc, 0.0f);
  }
}

// --------------------- WMMA fp32 GEMM: Y = act(X(+X2) @ W + b) -------------
// Block: 256 threads (8 waves); tile 128 rows x 128 cols; each wave owns a
// 16x128 strip (8 accumulator tiles). LDS holds operands in fragment-native
// layout so the hot loop is one ds_load_b64 per WMMA:
//   Af[kstep][wave][lane][j] = X[row0+16*wave+(lane&15)][4*kstep+2*(lane>>4)+j]
//   Wf[kstep][nt  ][lane][j] = W[4*kstep+2*(lane>>4)+j][16*nt+(lane&15)]
__global__ __launch_bounds__(256) void gemm_h_wmma_kernel(
    const float* __restrict__ X, const float* __restrict__ X2,
    const float* __restrict__ W, const float* __restrict__ bias,
    float* __restrict__ Y, int nRows, int doRelu) {
  extern __shared__ float smem[];
  float* Af = smem;           // 32 ksteps * 8 waves * 64 = 16384 floats (64 KB)
  float* Wf = smem + 16384;   // 32 ksteps * 8 ntiles * 64 = 16384 floats (64 KB)

  const int tid = threadIdx.x;
  const int lane = tid & 31;
  const int wave = tid >> 5;          // 0..7
  const int row0 = blockIdx.x * HD;   // first node row of this block tile

  // ---- stage X tile (optionally fused agg+h) into Af ----
  {
    const int r = tid >> 1;           // tile row 0..127
    const int cb = (tid & 1) * 64;    // column base 0 / 64
    const int gr = row0 + r;
    float* adst = Af + (r >> 4) * 128 + (r & 15) * 2;  // + kstep*1024 (+32 hi half)
    if (gr < nRows) {
      const float* xsrc = X + (long long)gr * HD + cb;
      const float* x2src = X2 ? (X2 + (long long)gr * HD + cb) : nullptr;
#pragma unroll
      for (int c = 0; c < 64; c += 4) {
        float4 v = *(const float4*)(xsrc + c);
        if (x2src) {
          float4 w = *(const float4*)(x2src + c);
          v.x += w.x; v.y += w.y; v.z += w.z; v.w += w.w;
        }
        const int kstep = (cb + c) >> 2;
        *(v2f*)(adst + kstep * 1024)      = (v2f){v.x, v.y};  // lane_half 0
        *(v2f*)(adst + kstep * 1024 + 32) = (v2f){v.z, v.w};  // lane_half 1
      }
    } else {
      const v2f z = (v2f){0.f, 0.f};
#pragma unroll
      for (int c = 0; c < 64; c += 4) {
        const int kstep = (cb + c) >> 2;
        *(v2f*)(adst + kstep * 1024)      = z;
        *(v2f*)(adst + kstep * 1024 + 32) = z;
      }
    }
  }
  // ---- stage W (128x128 row-major KxN) into Wf ----
  {
    const int q = tid >> 1;           // k-row pair index 0..127 -> rows 2q, 2q+1
    const int cb = (tid & 1) * 64;
    const int kstep = q >> 1;
    const int lh = q & 1;             // lane half (k%4)/2
    const float* w0 = W + (2 * q) * HD + cb;
    const float* w1 = w0 + HD;
    float* wdst = Wf + kstep * 512 + lh * 32;
#pragma unroll
    for (int c = 0; c < 64; c += 4) {
      float4 a = *(const float4*)(w0 + c);
      float4 b = *(const float4*)(w1 + c);
      const int n = cb + c;
      float* d = wdst + (n >> 4) * 64 + (n & 15) * 2;
      *(v2f*)(d + 0) = (v2f){a.x, b.x};
      *(v2f*)(d + 2) = (v2f){a.y, b.y};
      *(v2f*)(d + 4) = (v2f){a.z, b.z};
      *(v2f*)(d + 6) = (v2f){a.w, b.w};
    }
  }
  __syncthreads();

  // ---- WMMA main loop: 32 k-steps x 8 n-tiles ----
  v8f acc[8];
#pragma unroll
  for (int i = 0; i < 8; ++i)
    acc[i] = (v8f){0.f, 0.f, 0.f, 0.f, 0.f, 0.f, 0.f, 0.f};

  const float* abase = Af + wave * 128 + lane * 2;
  const float* bbase = Wf + lane * 2;

#pragma unroll 4
  for (int kstep = 0; kstep < 32; ++kstep) {
    v2f a = *(const v2f*)(abase + kstep * 1024);
    const float* bp = bbase + kstep * 512;
#pragma unroll
    for (int nt = 0; nt < 8; ++nt) {
      v2f b = *(const v2f*)(bp + nt * 64);
      acc[nt] = __builtin_amdgcn_wmma_f32_16x16x4_f32(
          false, a, false, b, (short)0, acc[nt], false, false);
    }
  }

  // ---- epilogue: bias + optional relu, direct store ----
  const int mlane = lane & 15;
  const int mbase = (lane >> 4) * 8;
  const int m0 = wave * 16;
#pragma unroll
  for (int nt = 0; nt < 8; ++nt) {
    const int col = nt * 16 + mlane;
    const float bv = bias[col];
#pragma unroll
    for (int j = 0; j < 8; ++j) {
      const int gr = row0 + m0 + mbase + j;
      if (gr < nRows) {
        float v = acc[nt][j] + bv;
        if (doRelu) v = fmaxf(v, 0.0f);
        Y[(long long)gr * HD + col] = v;
      }
    }
  }
}

// ----------------------------- pooling -------------------------------------
__global__ void pool_accum_kernel(const float* __restrict__ hbuf,
                                  const int* __restrict__ batch,
                                  float* __restrict__ sums,
                                  float* __restrict__ cnts, int n) {
  long long idx = (long long)blockIdx.x * blockDim.x + threadIdx.x;
  if (idx >= (long long)n * HD) return;
  int node = (int)(idx >> 7);
  int d = (int)(idx & (HD - 1));
  int g = batch[node];
  atomicAdd(&sums[(long long)g * HD + d], hbuf[idx]);
  if (d == 0) atomicAdd(&cnts[g], 1.0f);
}

__global__ void classify_kernel(const float* __restrict__ sums,
                                const float* __restrict__ cnts,
                                const float* __restrict__ Wc,
                                const float* __restrict__ bcv,
                                float* __restrict__ out, int Gn, int C) {
  int t = blockIdx.x * blockDim.x + threadIdx.x;
  if (t >= Gn * C) return;
  int g = t / C, c = t - g * C;
  float inv = 1.0f / fmaxf(cnts[g], 1.0f);
  float acc = bcv[c];
  for (int k = 0; k < HD; ++k) acc += (sums[(long long)g * HD + k] * inv) * Wc[k * C + c];
  out[t] = acc;
}

// ----------------------------- launcher ------------------------------------
extern "C" void kernel_launch(void* const* d_in, const int* in_sizes, int n_in,
                              void* d_out, int out_size, void* d_ws, size_t ws_size,
                              hipStream_t stream) {
  const float* x     = (const float*)d_in[0];
  const int*   src   = (const int*)d_in[1];
  const int*   dst   = (const int*)d_in[2];
  const int*   batch = (const int*)d_in[3];
  const float* p0    = (const float*)d_in[4];
  const float* prest = (const float*)d_in[5];
  const float* W1_0  = (const float*)d_in[6];
  const float* b1_0  = (const float*)d_in[7];
  const float* W2_0  = (const float*)d_in[8];
  const float* b2_0  = (const float*)d_in[9];
  const float* W1s   = (const float*)d_in[10];
  const float* b1s   = (const float*)d_in[11];
  const float* W2s   = (const float*)d_in[12];
  const float* b2s   = (const float*)d_in[13];
  const float* Wc    = (const float*)d_in[14];
  const float* bc    = (const float*)d_in[15];

  const int N  = in_sizes[3];          // batch vector length == #nodes
  const int E  = in_sizes[1];
  const int C  = in_sizes[15];         // bc length == #classes
  const int Gn = out_size / C;         // #graphs

  // workspace layout (floats)
  float* h    = (float*)d_ws;                 // N x 128
  float* agg  = h + (size_t)N * HD;           // N x 128 (N x 12 in layer 0)
  float* t    = agg + (size_t)N * HD;         // N x 128
  float* sums = t + (size_t)N * HD;           // Gn x 128
  float* cnts = sums + (size_t)Gn * HD;       // Gn

  const size_t shmem = (size_t)2 * 16384 * sizeof(float);  // 128 KB
  const int gemmBlocks = (N + HD - 1) / HD;

  // ---- layer 0 (D_IN = 12) ----
  zero_f32_kernel<<<1024, 256, 0, stream>>>(agg, (long long)N * 12);
  edge_scatter12_kernel<<<(E + 255) / 256, 256, 0, stream>>>(x, src, dst, p0, agg, E);
  mlp0_kernel<<<(N + 15) / 16, 128, 0, stream>>>(x, agg, W1_0, b1_0, t, N);
  gemm_h_wmma_kernel<<<gemmBlocks, 256, shmem, stream>>>(
      t, nullptr, W2_0, b2_0, h, N, /*relu=*/1);  // outer relu after layer 0

  // ---- layers 1..4 (H = 128) ----
  for (int l = 0; l < 4; ++l) {
    zero_f32_kernel<<<2048, 256, 0, stream>>>(agg, (long long)N * HD);
    edge_scatter128_kernel<<<(E + 7) / 8, 256, 0, stream>>>(
        h, src, dst, prest + l * HD, agg, E);
    gemm_h_wmma_kernel<<<gemmBlocks, 256, shmem, stream>>>(
        agg, h, W1s + l * HD * HD, b1s + l * HD, t, N, /*relu=*/1);
    gemm_h_wmma_kernel<<<gemmBlocks, 256, shmem, stream>>>(
        t, nullptr, W2s + l * HD * HD, b2s + l * HD, h, N, /*relu=*/(l < 3) ? 1 : 0);
  }

  // ---- mean pool + classifier ----
  zero_f32_kernel<<<64, 256, 0, stream>>>(sums, (long long)Gn * HD + Gn);
  pool_accum_kernel<<<(int)(((long long)N * HD + 255) / 256), 256, 0, stream>>>(
      h, batch, sums, cnts, N);
  classify_kernel<<<(Gn * C + 255) / 256, 256, 0, stream>>>(
      sums, cnts, Wc, bc, (float*)d_out, Gn, C);
}